// CenterPredictionLoss_59657095742069
// MI455X (gfx1250) — compile-verified
//
#include <hip/hip_runtime.h>
#include <math.h>

typedef __attribute__((ext_vector_type(2))) float v2f;
typedef __attribute__((ext_vector_type(8))) float v8f;

#define B_DIM 32
#define M_DIM 128
#define N_DIM 64
#define HINF 1e18f

__device__ __forceinline__ float wave_sum32(float x) {
    #pragma unroll
    for (int off = 16; off > 0; off >>= 1) x += __shfl_xor(x, off, 32);
    return x;
}

// One wave32 block per batch sample.
// Dynamic LDS layout (floats):
//   Csh   [128*128]  cost matrix (fp32)
//   ush   [129] vsh [129] minvsh [129]
//   predsh[128*3] tgtsh[64*2]
//   (ints) psh[129] waysh[129] usedsh[129] matched[128]
__global__ __launch_bounds__(32)
void cpl_match_loss_kernel(const float* __restrict__ pred,
                           const float* __restrict__ tgt,
                           float* __restrict__ partial)
{
    const int b    = blockIdx.x;
    const int lane = threadIdx.x;

    extern __shared__ char smem_raw[];
    float* Csh    = (float*)smem_raw;          // 16384
    float* ush    = Csh + M_DIM * M_DIM;       // 129
    float* vsh    = ush + (M_DIM + 1);         // 129
    float* minvsh = vsh + (M_DIM + 1);         // 129
    float* predsh = minvsh + (M_DIM + 1);      // 384
    float* tgtsh  = predsh + (M_DIM * 3);      // 128
    int*   psh    = (int*)(tgtsh + (N_DIM * 2));
    int*   waysh  = psh + (M_DIM + 1);
    int*   usedsh = waysh + (M_DIM + 1);
    int*   matched = usedsh + (M_DIM + 1);     // 128

    // ---- stage inputs into LDS ----
    for (int i = lane; i < M_DIM * 3; i += 32) predsh[i] = pred[b * M_DIM * 3 + i];
    for (int i = lane; i < N_DIM * 2; i += 32) tgtsh[i]  = tgt[b * N_DIM * 2 + i];
    __syncthreads();

    // ---- build cost matrix with V_WMMA_F32_16X16X4_F32 ----
    // d^2 = A(16x4) x B(4x16) with A row m = [px, py, px^2+py^2, 1],
    //                               B col n = [-2tx, -2ty, 1, tx^2+ty^2]
    const int hl  = lane >> 4;   // 0: K0/K1 half, 1: K2/K3 half
    const int l15 = lane & 15;

    for (int ti = 0; ti < 8; ++ti) {
        const int m = ti * 16 + l15;
        const float px = predsh[m * 3 + 0];
        const float py = predsh[m * 3 + 1];
        v2f a;
        a.x = hl ? (px * px + py * py) : px;
        a.y = hl ? 1.0f : py;
        for (int tj = 0; tj < 8; ++tj) {
            const int n = tj * 16 + l15;
            v2f bv;
            if (n < N_DIM) {
                const float tx = tgtsh[n * 2 + 0];
                const float ty = tgtsh[n * 2 + 1];
                bv.x = hl ? 1.0f : (-2.0f * tx);
                bv.y = hl ? (tx * tx + ty * ty) : (-2.0f * ty);
            } else {
                bv.x = 0.0f; bv.y = 0.0f;
            }
            v8f acc = {0.f, 0.f, 0.f, 0.f, 0.f, 0.f, 0.f, 0.f};
            acc = __builtin_amdgcn_wmma_f32_16x16x4_f32(
                false, a, false, bv, (short)0, acc, false, false);
            // D layout: VGPR r -> row (r + 8*hl), col = l15
            #pragma unroll
            for (int r = 0; r < 8; ++r) {
                const int mm = ti * 16 + r + hl * 8;
                const int nn = tj * 16 + l15;
                float c;
                if (nn < N_DIM) {
                    const float d2 = fmaxf(acc[r], 0.0f);
                    c = sqrtf(d2) - 0.1f * predsh[mm * 3 + 2];
                } else {
                    c = 0.0f;   // zero padding, matches reference sq matrix
                }
                Csh[mm * M_DIM + nn] = c;
            }
        }
    }
    __syncthreads();

    // ---- Hungarian (e-maxx potentials), lane-parallel inner loops ----
    const int n1 = M_DIM + 1; // 129
    for (int j = lane; j < n1; j += 32) {
        ush[j] = 0.0f; vsh[j] = 0.0f; psh[j] = 0; waysh[j] = 0;
    }
    __syncthreads();

    for (int i = 1; i <= M_DIM; ++i) {
        if (lane == 0) psh[0] = i;
        for (int j = lane; j < n1; j += 32) { minvsh[j] = HINF; usedsh[j] = 0; }
        __syncthreads();

        int j0 = 0;
        for (;;) {
            if (lane == 0) usedsh[j0] = 1;
            __syncthreads();
            const int   i0  = psh[j0];
            const float ui0 = ush[i0];

            // relax unused columns against row i0
            for (int j = lane; j < n1; j += 32) {
                if (!usedsh[j] && j >= 1) {
                    const float cur = Csh[(i0 - 1) * M_DIM + (j - 1)] - ui0 - vsh[j];
                    if (cur < minvsh[j]) { minvsh[j] = cur; waysh[j] = j0; }
                }
            }
            __syncthreads();

            // masked argmin (np.argmin tie-break: lowest j)
            float bestv = HINF;
            int   bestj = 0;
            for (int j = lane; j < n1; j += 32) {
                const float mv = usedsh[j] ? HINF : minvsh[j];
                if (mv < bestv) { bestv = mv; bestj = j; }
            }
            #pragma unroll
            for (int off = 16; off > 0; off >>= 1) {
                const float ov = __shfl_xor(bestv, off, 32);
                const int   oj = __shfl_xor(bestj, off, 32);
                if (ov < bestv || (ov == bestv && oj < bestj)) { bestv = ov; bestj = oj; }
            }
            const int   j1    = bestj;
            const float delta = bestv;

            // update potentials; p[j] values are distinct over used j -> race-free
            for (int j = lane; j < n1; j += 32) {
                if (usedsh[j]) { ush[psh[j]] += delta; vsh[j] -= delta; }
                else           { minvsh[j] -= delta; }
            }
            __syncthreads();

            j0 = j1;
            if (psh[j0] == 0) break;
        }
        // augment alternating path (serial, lane 0)
        if (lane == 0) {
            int jj = j0;
            while (jj) {
                const int jprev = waysh[jj];
                psh[jj] = psh[jprev];
                jj = jprev;
            }
        }
        __syncthreads();
    }

    // ---- loss for this sample ----
    for (int m = lane; m < M_DIM; m += 32) matched[m] = 0;
    __syncthreads();

    float coord = 0.0f, bcem = 0.0f;
    for (int nidx = lane; nidx < N_DIM; nidx += 32) {
        const int idx = psh[nidx + 1] - 1;   // pred index assigned to GT column nidx
        matched[idx] = 1;                    // indices unique -> no race
        const float cx = predsh[idx * 3 + 0];
        const float cy = predsh[idx * 3 + 1];
        const float cf = predsh[idx * 3 + 2];
        const float tx = tgtsh[nidx * 2 + 0];
        const float ty = tgtsh[nidx * 2 + 1];
        coord += (cx - tx) * (cx - tx) + (cy - ty) * (cy - ty);
        bcem  += -fmaxf(logf(cf), -100.0f);
    }
    __syncthreads();

    float bceu = 0.0f;
    for (int m = lane; m < M_DIM; m += 32) {
        if (!matched[m]) {
            const float cf = predsh[m * 3 + 2];
            bceu += -fmaxf(logf(1.0f - cf), -100.0f);
        }
    }

    coord = wave_sum32(coord);
    bcem  = wave_sum32(bcem);
    bceu  = wave_sum32(bceu);

    if (lane == 0) {
        const float coord_b = coord / (float)(N_DIM * 2);
        const float conf_b  = bcem / (float)N_DIM + bceu / (float)(M_DIM - N_DIM);
        partial[b] = coord_b + conf_b;
    }
}

__global__ void cpl_reduce_kernel(const float* __restrict__ partial,
                                  float* __restrict__ out)
{
    if (threadIdx.x == 0 && blockIdx.x == 0) {
        float s = 0.0f;
        for (int b = 0; b < B_DIM; ++b) s += partial[b];  // fixed order: deterministic
        out[0] = s / (float)B_DIM;
    }
}

extern "C" void kernel_launch(void* const* d_in, const int* in_sizes, int n_in,
                              void* d_out, int out_size, void* d_ws, size_t ws_size,
                              hipStream_t stream) {
    (void)in_sizes; (void)n_in; (void)out_size; (void)ws_size;
    const float* pred = (const float*)d_in[0];   // (32,128,3) f32
    const float* tgt  = (const float*)d_in[1];   // (32,64,2)  f32
    float* out        = (float*)d_out;           // scalar f32
    float* partial    = (float*)d_ws;            // 32 floats of scratch

    // dynamic LDS: 16384 + 3*129 + 384 + 128 floats, 3*129 + 128 ints = 71192 B
    const size_t shbytes = (size_t)(16384 + 3 * 129 + 384 + 128) * 4
                         + (size_t)(3 * 129 + 128) * 4;

    cpl_match_loss_kernel<<<B_DIM, 32, shbytes, stream>>>(pred, tgt, partial);
    cpl_reduce_kernel<<<1, 32, 0, stream>>>(partial, out);
}